// KNeighborsModel_7241314861221
// MI455X (gfx1250) — compile-verified
//
#include <hip/hip_runtime.h>
#include <hip/hip_bf16.h>
#include <math.h>

#define NTRAIN 100000
#define NQUERY 1024
#define KDIM   64
#define TOPK   16
#define NCLS   10
#define ROWS   16            // query rows per block (one WMMA M tile)
#define WAVES  4
#define NTHREADS (WAVES * 32)
#define CHUNK  (WAVES * 16)  // 64 train columns per chunk
#define TPR    (NTHREADS / ROWS) // 8 selection threads per row
#define SPLIT  8             // train set split across blockIdx.y
#define CH_STRIDE 68         // padded LDS row stride (floats)

typedef float v2f __attribute__((ext_vector_type(2)));
typedef float v8f __attribute__((ext_vector_type(8)));

// ---------------- Kernel 1: per-train-row squared norms ----------------
__global__ void knn_t2_kernel(const float* __restrict__ train,
                              float* __restrict__ t2) {
  int j = blockIdx.x * blockDim.x + threadIdx.x;
  if (j >= NTRAIN) return;
  const float4* row = (const float4*)(train + (size_t)j * KDIM);
  float s = 0.f;
#pragma unroll
  for (int i = 0; i < KDIM / 4; ++i) {
    float4 v = row[i];
    s += v.x * v.x + v.y * v.y + v.z * v.z + v.w * v.w;
  }
  t2[j] = s;
}

// ---------------- Kernel 2: fused WMMA distances + top-16 ----------------
__launch_bounds__(NTHREADS)
__global__ void knn_main_kernel(const float* __restrict__ x,
                                const float* __restrict__ train,
                                const float* __restrict__ t2g,
                                float* __restrict__ partD,
                                int*   __restrict__ partI) {
  __shared__ float xs[ROWS * KDIM];                  // 4 KB query tile
  __shared__ float chunkLDS[ROWS * CH_STRIDE];       // 16x64 d2 tile (padded)
  __shared__ float candD[ROWS * TPR * TOPK];         // per-thread top-16 values
  __shared__ int   candI[ROWS * TPR * TOPK];         // per-thread top-16 indices
  __shared__ float x2s[ROWS];

  const int tid   = threadIdx.x;
  const int lane  = tid & 31;
  const int wave  = tid >> 5;
  const int part  = blockIdx.y;
  const int rowBase = blockIdx.x * ROWS;
  const float INF = __builtin_inff();

  // ---- load 16x64 query tile (128 threads * 2 float4) ----
  {
    const float4* src = (const float4*)(x + (size_t)rowBase * KDIM);
    float4* dst = (float4*)xs;
    dst[tid]            = src[tid];
    dst[tid + NTHREADS] = src[tid + NTHREADS];
  }
  __syncthreads();

  if (tid < ROWS) {
    float s = 0.f;
#pragma unroll
    for (int k = 0; k < KDIM; ++k) { float v = xs[tid * KDIM + k]; s += v * v; }
    x2s[tid] = s;
  }
  for (int i = tid; i < ROWS * TPR * TOPK; i += NTHREADS) {
    candD[i] = INF; candI[i] = 0;
  }
  __syncthreads();

  // ---- preload A fragments (constant across all chunks) ----
  const int frow = lane & 15;            // M for A, N for B
  const int koff = (lane >> 4) * 2;      // K offset 0 or 2 per ISA layout
  v2f afrag[16];
#pragma unroll
  for (int kk = 0; kk < 16; ++kk) {
    afrag[kk].x = xs[frow * KDIM + kk * 4 + koff];
    afrag[kk].y = xs[frow * KDIM + kk * 4 + koff + 1];
  }
  const int mhi = (lane >> 4) * 8;
  float xm[8];
#pragma unroll
  for (int v = 0; v < 8; ++v) xm[v] = x2s[v + mhi];

  // ---- per-thread selection state (lists live in LDS, threshold in VGPR) ----
  const int selRow = tid >> 3;           // 0..15
  const int selSub = tid & 7;            // 0..7
  float* myD = &candD[(selRow * TPR + selSub) * TOPK];
  int*   myI = &candI[(selRow * TPR + selSub) * TOPK];
  float worst = INF;

  const int nChunks = (NTRAIN + CHUNK - 1) / CHUNK;           // 1563
  const int perPart = (nChunks + SPLIT - 1) / SPLIT;          // 196
  const int chBeg = part * perPart;
  const int chEndRaw = chBeg + perPart;
  const int chEnd = (chEndRaw < nChunks) ? chEndRaw : nChunks;

  for (int ch = chBeg; ch < chEnd; ++ch) {
    const int nBase = ch * CHUNK;
    const int col   = nBase + wave * 16 + frow;
    const int colC  = (col < NTRAIN) ? col : (NTRAIN - 1);
    const float* brow = train + (size_t)colC * KDIM;

    // 4 independent accumulators: WMMA RAW chain depth 16 -> 4
    v8f acc0 = {0.f,0.f,0.f,0.f,0.f,0.f,0.f,0.f};
    v8f acc1 = {0.f,0.f,0.f,0.f,0.f,0.f,0.f,0.f};
    v8f acc2 = {0.f,0.f,0.f,0.f,0.f,0.f,0.f,0.f};
    v8f acc3 = {0.f,0.f,0.f,0.f,0.f,0.f,0.f,0.f};
#pragma unroll
    for (int kk = 0; kk < 16; kk += 4) {
      v2f b0, b1, b2, b3;
      b0.x = brow[(kk+0) * 4 + koff]; b0.y = brow[(kk+0) * 4 + koff + 1];
      b1.x = brow[(kk+1) * 4 + koff]; b1.y = brow[(kk+1) * 4 + koff + 1];
      b2.x = brow[(kk+2) * 4 + koff]; b2.y = brow[(kk+2) * 4 + koff + 1];
      b3.x = brow[(kk+3) * 4 + koff]; b3.y = brow[(kk+3) * 4 + koff + 1];
      acc0 = __builtin_amdgcn_wmma_f32_16x16x4_f32(false, afrag[kk+0], false, b0,
                                                   (short)0, acc0, false, false);
      acc1 = __builtin_amdgcn_wmma_f32_16x16x4_f32(false, afrag[kk+1], false, b1,
                                                   (short)0, acc1, false, false);
      acc2 = __builtin_amdgcn_wmma_f32_16x16x4_f32(false, afrag[kk+2], false, b2,
                                                   (short)0, acc2, false, false);
      acc3 = __builtin_amdgcn_wmma_f32_16x16x4_f32(false, afrag[kk+3], false, b3,
                                                   (short)0, acc3, false, false);
    }
    v8f acc = (acc0 + acc1) + (acc2 + acc3);

    // unconditional (clamped) t2 load; branchless OOB select
    const float t2c = t2g[colC];
    const bool inRange = (col < NTRAIN);
#pragma unroll
    for (int v = 0; v < 8; ++v) {
      const int M = v + mhi;
      const float d2v = xm[v] + t2c - 2.0f * acc[v];
      chunkLDS[M * CH_STRIDE + wave * 16 + frow] = inRange ? d2v : INF;
    }
    __syncthreads();

    // selection: each thread scans 8 columns of its row
#pragma unroll
    for (int m = 0; m < 8; ++m) {
      const int c = selSub + m * 8;
      const float v = chunkLDS[selRow * CH_STRIDE + c];
      if (v < worst) {
        float mx = myD[0]; int mxs = 0;
#pragma unroll
        for (int s2 = 1; s2 < TOPK; ++s2) {
          float dv = myD[s2];
          if (dv > mx) { mx = dv; mxs = s2; }
        }
        myD[mxs] = v; myI[mxs] = nBase + c;
        float nmx = myD[0];
#pragma unroll
        for (int s2 = 1; s2 < TOPK; ++s2) {
          float dv = myD[s2];
          if (dv > nmx) nmx = dv;
        }
        worst = nmx;
      }
    }
    __syncthreads();
  }

  // ---- per-row 128 -> 16 merge (ascending order), write partial lists ----
  if (tid < ROWS) {
    float* rD = &candD[tid * TPR * TOPK];
    int*   rI = &candI[tid * TPR * TOPK];
    const int gr = rowBase + tid;
    float* pD = partD + ((size_t)(part * NQUERY + gr)) * TOPK;
    int*   pI = partI + ((size_t)(part * NQUERY + gr)) * TOPK;
#pragma unroll
    for (int t = 0; t < TOPK; ++t) {
      float best = INF; int bs = 0;
      for (int c = 0; c < TPR * TOPK; ++c) {
        float dv = rD[c];
        if (dv < best) { best = dv; bs = c; }
      }
      pD[t] = best;
      pI[t] = rI[bs];
      rD[bs] = INF;
    }
  }
}

// ---------------- Kernel 3: SPLIT-way merge + weighted vote + argmax ----------------
__global__ void knn_final_kernel(const float* __restrict__ partD,
                                 const int*   __restrict__ partI,
                                 const int*   __restrict__ labels,
                                 float* __restrict__ out) {
  const int row = blockIdx.x * blockDim.x + threadIdx.x;
  if (row >= NQUERY) return;
  const float INF = __builtin_inff();

  const float* Ld[SPLIT];
  const int*   Li[SPLIT];
  int   p[SPLIT];
  float h[SPLIT];
#pragma unroll
  for (int q = 0; q < SPLIT; ++q) {
    Ld[q] = partD + ((size_t)(q * NQUERY + row)) * TOPK;
    Li[q] = partI + ((size_t)(q * NQUERY + row)) * TOPK;
    p[q] = 0;
    h[q] = Ld[q][0];
  }

  float nd[TOPK]; int ni[TOPK];
#pragma unroll
  for (int t = 0; t < TOPK; ++t) {
    int q = 0; float best = h[0];
#pragma unroll
    for (int j = 1; j < SPLIT; ++j) {
      if (h[j] < best) { best = h[j]; q = j; }
    }
    nd[t] = best;
    ni[t] = Li[q][p[q]];
    p[q]++;
    h[q] = (p[q] < TOPK) ? Ld[q][p[q]] : INF;
  }

  // distances, inverse-distance weights with exact-match handling
  bool hasZero = false;
#pragma unroll
  for (int t = 0; t < TOPK; ++t) {
    float d2v = nd[t] < 0.f ? 0.f : nd[t];
    float dd = sqrtf(d2v);
    nd[t] = dd;
    if (dd == 0.f) hasZero = true;
  }
  float w[TOPK];
#pragma unroll
  for (int t = 0; t < TOPK; ++t) {
    w[t] = hasZero ? ((nd[t] == 0.f) ? 1.f : 0.f) : (1.f / nd[t]);
  }

  float proba[NCLS];
#pragma unroll
  for (int c = 0; c < NCLS; ++c) proba[c] = 0.f;
#pragma unroll
  for (int t = 0; t < TOPK; ++t) {
    const int lab = labels[ni[t]];
#pragma unroll
    for (int c = 0; c < NCLS; ++c) proba[c] += (lab == c) ? w[t] : 0.f;
  }
  float s = 0.f;
#pragma unroll
  for (int c = 0; c < NCLS; ++c) s += proba[c];
  if (s == 0.f) s = 1.f;
  const float inv = 1.f / s;
  int arg = 0; float mx = -1.f;
#pragma unroll
  for (int c = 0; c < NCLS; ++c) {
    proba[c] *= inv;
    if (proba[c] > mx) { mx = proba[c]; arg = c; }
  }

  out[row] = (float)arg;
  float* pr = out + NQUERY + (size_t)row * NCLS;
#pragma unroll
  for (int c = 0; c < NCLS; ++c) pr[c] = proba[c];
}

// ---------------- launch ----------------
extern "C" void kernel_launch(void* const* d_in, const int* in_sizes, int n_in,
                              void* d_out, int out_size, void* d_ws, size_t ws_size,
                              hipStream_t stream) {
  const float* x      = (const float*)d_in[0];
  const float* train  = (const float*)d_in[1];
  const int*   labels = (const int*)d_in[2];
  float* out = (float*)d_out;

  // workspace layout: t2[100000] | partD[SPLIT*1024*16] f32 | partI[SPLIT*1024*16] i32
  float* t2    = (float*)d_ws;
  float* partD = t2 + NTRAIN;
  int*   partI = (int*)(partD + (size_t)SPLIT * NQUERY * TOPK);

  knn_t2_kernel<<<(NTRAIN + 255) / 256, 256, 0, stream>>>(train, t2);

  dim3 grid2(NQUERY / ROWS, SPLIT);  // 64 x 8 = 512 workgroups
  knn_main_kernel<<<grid2, NTHREADS, 0, stream>>>(x, train, t2, partD, partI);

  knn_final_kernel<<<(NQUERY + 127) / 128, 128, 0, stream>>>(partD, partI, labels, out);
}